// TemporalSSMEfficient_2173253452153
// MI455X (gfx1250) — compile-verified
//
#include <hip/hip_runtime.h>
#include <hip/hip_bf16.h>
#include <math.h>

#define BB 16
#define TT 15
#define HH 480
#define WW 640
#define CH 16
#define PP 8
#define BLKH 30   // H/16
#define BLKW 40   // W/16

typedef float v2f __attribute__((ext_vector_type(2)));
typedef float v8f __attribute__((ext_vector_type(8)));

__device__ __forceinline__ float gelu_exact(float v) {
    return 0.5f * v * (1.0f + erff(v * 0.70710678118654752f));
}

// ---------------------------------------------------------------------------
// Kernel 1: block-average downsample  x(B,T,480,640) -> xd(B,T,16,16)
// One block per (b,t) frame; thread (d1,d2) averages its 30x40 tile (float4).
// ---------------------------------------------------------------------------
__global__ __launch_bounds__(256) void k_downsample(const float* __restrict__ x,
                                                    float* __restrict__ xd) {
    const int bt = blockIdx.x;                 // 0..239
    const int d1 = threadIdx.x >> 4;
    const int d2 = threadIdx.x & 15;
    const float* base = x + ((size_t)bt * HH + (size_t)d1 * BLKH) * WW + d2 * BLKW;
    float s = 0.0f;
    for (int r = 0; r < BLKH; ++r) {
        const float4* row = (const float4*)(base + (size_t)r * WW);
        __builtin_prefetch(base + (size_t)(r + 2) * WW, 0, 0);  // global_prefetch_b8
#pragma unroll
        for (int q = 0; q < BLKW / 4; ++q) {
            float4 v = row[q];
            s += v.x + v.y + v.z + v.w;
        }
    }
    xd[(size_t)bt * 256 + threadIdx.x] = s * (1.0f / (BLKH * BLKW));
}

// ---------------------------------------------------------------------------
// Kernel 2: SSM recurrence + GLU MLP with f32 WMMA (16x16x4).
// 32 blocks x 8 waves; each wave owns a 16-sequence tile (4096 seqs total).
// Lane l computes sequence (l & 15); upper half duplicates -> EXEC all ones.
// ---------------------------------------------------------------------------

// A in registers (row-per-lane), B from LDS weight array W[n][k].
#define WMMA_REG_CHUNK(acc, FXA, W, nrow, k0)                                   \
    {                                                                           \
        v2f a_, b_;                                                             \
        a_.x = hi ? FXA[(k0) + 2] : FXA[(k0)];                                  \
        a_.y = hi ? FXA[(k0) + 3] : FXA[(k0) + 1];                              \
        int kb_ = (k0) + (hi ? 2 : 0);                                          \
        b_.x = W[nrow][kb_]; b_.y = W[nrow][kb_ + 1];                           \
        acc = __builtin_amdgcn_wmma_f32_16x16x4_f32(false, a_, false, b_,       \
                                                    (short)0, acc, false, false); \
    }

// A from LDS transpose scratch AL[m][k].
#define WMMA_LDS_CHUNK(acc, AL, W, nrow, k0)                                    \
    {                                                                           \
        v2f a_, b_;                                                             \
        int kb_ = (k0) + (hi ? 2 : 0);                                          \
        a_.x = AL[m16][kb_]; a_.y = AL[m16][kb_ + 1];                           \
        b_.x = W[nrow][kb_]; b_.y = W[nrow][kb_ + 1];                           \
        acc = __builtin_amdgcn_wmma_f32_16x16x4_f32(false, a_, false, b_,       \
                                                    (short)0, acc, false, false); \
    }

__global__ __launch_bounds__(256) void k_ssm_mlp(
    const float* __restrict__ xd,
    const float* __restrict__ w_in, const float* __restrict__ b_in,
    const float* __restrict__ ln1_g, const float* __restrict__ ln1_b,
    const float* __restrict__ Lre, const float* __restrict__ Lim,
    const float* __restrict__ B_re, const float* __restrict__ B_im,
    const float* __restrict__ C_re, const float* __restrict__ C_im,
    const float* __restrict__ Dv, const float* __restrict__ log_step,
    const float* __restrict__ ln2_g, const float* __restrict__ ln2_b,
    const float* __restrict__ w_enc, const float* __restrict__ w_dec,
    const float* __restrict__ w_out, const float* __restrict__ b_out,
    float* __restrict__ yd) {

    __shared__ float s_win[CH], s_bin[CH], s_l1g[CH], s_l1b[CH];
    __shared__ float s_D[CH], s_l2g[CH], s_l2b[CH], s_bout[CH];
    __shared__ float s_lr[PP], s_li[PP], s_cr[PP], s_ci[PP], s_msk[PP];
    __shared__ float s_Bbr[PP][CH], s_Bbi[PP][CH];
    __shared__ float s_Ctr[CH][PP], s_Cti[CH][PP];
    __shared__ float s_wenc[2 * CH][CH];
    __shared__ float s_wdec[CH][CH];
    __shared__ float s_wout[CH][CH];
    __shared__ float s_tA[8][16][17];   // WMMA D->A transpose scratch (padded)
    __shared__ float s_tB[8][16][17];   // fx2 residual in D layout

    const int tid = threadIdx.x;

    // ---- stage small parameters into LDS -----------------------------------
    if (tid < CH) {
        s_win[tid] = w_in[tid];  s_bin[tid] = b_in[tid];
        s_l1g[tid] = ln1_g[tid]; s_l1b[tid] = ln1_b[tid];
        s_D[tid]   = Dv[tid];
        s_l2g[tid] = ln2_g[tid]; s_l2b[tid] = ln2_b[tid];
        s_bout[tid] = b_out[tid];
    }
    if (tid < PP) {
        float st = expf(log_step[tid]);
        float lr = Lre[tid], li = Lim[tid];
        float er = expf(lr * st);
        float lbr = er * cosf(li * st);
        float lbi = er * sinf(li * st);
        s_lr[tid] = lbr; s_li[tid] = lbi;
        // (Lam_bar - 1)/Lam  via complex division
        float inv = 1.0f / (lr * lr + li * li);
        float nr = lbr - 1.0f, ni = lbi;
        s_cr[tid] = (nr * lr + ni * li) * inv;
        s_ci[tid] = (ni * lr - nr * li) * inv;
        float freq = st * fabsf(li) * (1.0f / (2.0f * 3.14159265358979323846f));
        s_msk[tid] = (freq < 0.25f) ? 1.0f : 0.0f;
    }
    for (int it = tid; it < 2 * CH * CH; it += 256)
        s_wenc[it >> 4][it & 15] = w_enc[it];
    s_wdec[tid >> 4][tid & 15] = w_dec[tid];
    s_wout[tid >> 4][tid & 15] = w_out[tid];
    __syncthreads();
    if (tid < PP * CH) {
        int p = tid >> 4, c = tid & 15;
        float br = B_re[p * CH + c], bi = B_im[p * CH + c];
        s_Bbr[p][c] = s_cr[p] * br - s_ci[p] * bi;
        s_Bbi[p][c] = s_cr[p] * bi + s_ci[p] * br;
    } else if (tid < 2 * PP * CH) {
        int k = tid - PP * CH;
        int c = k >> 3, p = k & 7;
        s_Ctr[c][p] = C_re[c * PP + p] * s_msk[p];
        s_Cti[c][p] = C_im[c * PP + p] * s_msk[p];
    }
    __syncthreads();

    // ---- per-lane scalar SSM -----------------------------------------------
    const int wave = tid >> 5;
    const int lane = tid & 31;
    const int m16  = lane & 15;
    const bool hi  = (lane >= 16);
    const int tile = blockIdx.x * 8 + wave;     // 0..255
    const int seq  = tile * 16 + m16;           // 0..4095 (dup in upper half)
    const int pb   = seq >> 8;
    const int pd1  = (seq >> 4) & 15;
    const int pd2  = seq & 15;

    float sr[PP], si[PP], fxl[CH];
#pragma unroll
    for (int p = 0; p < PP; ++p) { sr[p] = 0.0f; si[p] = 0.0f; }

    for (int t = 0; t < TT; ++t) {
        float xv = xd[(((size_t)pb * TT + t) * 16 + pd1) * 16 + pd2];
        float u[CH], fx[CH];
        float mean = 0.0f;
#pragma unroll
        for (int c = 0; c < CH; ++c) { u[c] = xv * s_win[c] + s_bin[c]; mean += u[c]; }
        mean *= (1.0f / CH);
        float var = 0.0f;
#pragma unroll
        for (int c = 0; c < CH; ++c) { float d = u[c] - mean; var += d * d; }
        var *= (1.0f / CH);
        float inv = rsqrtf(var + 1e-5f);
#pragma unroll
        for (int c = 0; c < CH; ++c)
            fx[c] = (u[c] - mean) * inv * s_l1g[c] + s_l1b[c];
#pragma unroll
        for (int p = 0; p < PP; ++p) {
            float br = 0.0f, bi = 0.0f;
#pragma unroll
            for (int c = 0; c < CH; ++c) { br += fx[c] * s_Bbr[p][c]; bi += fx[c] * s_Bbi[p][c]; }
            float nr = s_lr[p] * sr[p] - s_li[p] * si[p] + br;
            float ni = s_lr[p] * si[p] + s_li[p] * sr[p] + bi;
            sr[p] = nr; si[p] = ni;
        }
        if (t == TT - 1) {
#pragma unroll
            for (int c = 0; c < CH; ++c) fxl[c] = fx[c];
        }
    }

    // ---- last-step readout + GELU residual + LN2 ---------------------------
    float x1[CH], fx2r[CH];
    float m2 = 0.0f;
#pragma unroll
    for (int c = 0; c < CH; ++c) {
        float ys = fxl[c] * s_D[c];
#pragma unroll
        for (int p = 0; p < PP; ++p)
            ys += sr[p] * s_Ctr[c][p] - si[p] * s_Cti[c][p];
        x1[c] = gelu_exact(ys) + fxl[c];
        m2 += x1[c];
    }
    m2 *= (1.0f / CH);
    float v2 = 0.0f;
#pragma unroll
    for (int c = 0; c < CH; ++c) { float d = x1[c] - m2; v2 += d * d; }
    v2 *= (1.0f / CH);
    float inv2 = rsqrtf(v2 + 1e-5f);
#pragma unroll
    for (int c = 0; c < CH; ++c)
        fx2r[c] = (x1[c] - m2) * inv2 * s_l2g[c] + s_l2b[c];

    // ---- WMMA GLU MLP: e = fx2 @ Wenc^T ------------------------------------
    v8f e1 = {0.f,0.f,0.f,0.f,0.f,0.f,0.f,0.f};
    v8f e2 = {0.f,0.f,0.f,0.f,0.f,0.f,0.f,0.f};
    WMMA_REG_CHUNK(e1, fx2r, s_wenc, m16, 0);
    WMMA_REG_CHUNK(e1, fx2r, s_wenc, m16, 4);
    WMMA_REG_CHUNK(e1, fx2r, s_wenc, m16, 8);
    WMMA_REG_CHUNK(e1, fx2r, s_wenc, m16, 12);
    WMMA_REG_CHUNK(e2, fx2r, s_wenc, 16 + m16, 0);
    WMMA_REG_CHUNK(e2, fx2r, s_wenc, 16 + m16, 4);
    WMMA_REG_CHUNK(e2, fx2r, s_wenc, 16 + m16, 8);
    WMMA_REG_CHUNK(e2, fx2r, s_wenc, 16 + m16, 12);

    // GLU elementwise (layout-invariant), stash h and fx2 for next GEMM
    const int mrow = hi ? 8 : 0;
#pragma unroll
    for (int j = 0; j < 8; ++j)
        s_tA[wave][mrow + j][m16] = e1[j] * gelu_exact(e2[j]);
    if (!hi) {
#pragma unroll
        for (int c = 0; c < CH; ++c) s_tB[wave][m16][c] = fx2r[c];
    }
    __syncthreads();

    // x2 = h @ Wdec^T + fx2  (fx2 injected as C operand in D layout)
    v8f x2f;
#pragma unroll
    for (int j = 0; j < 8; ++j) x2f[j] = s_tB[wave][mrow + j][m16];
    WMMA_LDS_CHUNK(x2f, s_tA[wave], s_wdec, m16, 0);
    WMMA_LDS_CHUNK(x2f, s_tA[wave], s_wdec, m16, 4);
    WMMA_LDS_CHUNK(x2f, s_tA[wave], s_wdec, m16, 8);
    WMMA_LDS_CHUNK(x2f, s_tA[wave], s_wdec, m16, 12);
    __syncthreads();
#pragma unroll
    for (int j = 0; j < 8; ++j) s_tA[wave][mrow + j][m16] = x2f[j];
    __syncthreads();

    // agg = x2 @ Wout^T + b_out
    v8f af;
#pragma unroll
    for (int j = 0; j < 8; ++j) af[j] = s_bout[m16];
    WMMA_LDS_CHUNK(af, s_tA[wave], s_wout, m16, 0);
    WMMA_LDS_CHUNK(af, s_tA[wave], s_wout, m16, 4);
    WMMA_LDS_CHUNK(af, s_tA[wave], s_wout, m16, 8);
    WMMA_LDS_CHUNK(af, s_tA[wave], s_wout, m16, 12);

    // Scatter directly into yd(b, CH, 16, 16); n = channel = m16, m = seq row
#pragma unroll
    for (int j = 0; j < 8; ++j) {
        int m  = mrow + j;
        int sq = tile * 16 + m;
        int ob = sq >> 8, od1 = (sq >> 4) & 15, od2 = sq & 15;
        yd[(((size_t)ob * CH + m16) * 16 + od1) * 16 + od2] = af[j];
    }
}

// ---------------------------------------------------------------------------
// Kernel 3: bilinear upsample yd(B,CH,16,16) -> out(B,CH,480,640).
// One thread per float4 of output; yd (256 KB) stays L2-resident.
// ---------------------------------------------------------------------------
__global__ __launch_bounds__(256) void k_upsample(const float* __restrict__ yd,
                                                  float* __restrict__ out) {
    const int total4 = BB * CH * HH * (WW / 4);
    int idx = blockIdx.x * blockDim.x + threadIdx.x;
    if (idx >= total4) return;
    int j4    = idx % (WW / 4);
    int rest  = idx / (WW / 4);
    int i     = rest % HH;
    int plane = rest / HH;                     // b*CH + c

    float ry = (float)i * (15.0f / 479.0f);
    int y0 = (int)ry;
    int y1 = min(y0 + 1, 15);
    float wy = ry - (float)y0;
    const float* p0 = yd + (size_t)plane * 256 + y0 * 16;
    const float* p1 = yd + (size_t)plane * 256 + y1 * 16;

    float4 res;
    float* r = (float*)&res;
#pragma unroll
    for (int q = 0; q < 4; ++q) {
        int j = j4 * 4 + q;
        float rx = (float)j * (15.0f / 639.0f);
        int x0 = (int)rx;
        int x1 = min(x0 + 1, 15);
        float wx = rx - (float)x0;
        float top = p0[x0] * (1.0f - wx) + p0[x1] * wx;
        float bot = p1[x0] * (1.0f - wx) + p1[x1] * wx;
        r[q] = top * (1.0f - wy) + bot * wy;
    }
    *(float4*)(out + (size_t)plane * (HH * WW) + (size_t)i * WW + j4 * 4) = res;
}

// ---------------------------------------------------------------------------
extern "C" void kernel_launch(void* const* d_in, const int* in_sizes, int n_in,
                              void* d_out, int out_size, void* d_ws, size_t ws_size,
                              hipStream_t stream) {
    const float* x        = (const float*)d_in[0];
    const float* w_in     = (const float*)d_in[1];
    const float* b_in     = (const float*)d_in[2];
    const float* ln1_g    = (const float*)d_in[3];
    const float* ln1_b    = (const float*)d_in[4];
    const float* Lre      = (const float*)d_in[5];
    const float* Lim      = (const float*)d_in[6];
    const float* B_re     = (const float*)d_in[7];
    const float* B_im     = (const float*)d_in[8];
    const float* C_re     = (const float*)d_in[9];
    const float* C_im     = (const float*)d_in[10];
    const float* Dv       = (const float*)d_in[11];
    const float* log_step = (const float*)d_in[12];
    const float* ln2_g    = (const float*)d_in[13];
    const float* ln2_b    = (const float*)d_in[14];
    const float* w_enc    = (const float*)d_in[15];
    const float* w_dec    = (const float*)d_in[16];
    const float* w_out    = (const float*)d_in[17];
    const float* b_out    = (const float*)d_in[18];

    float* ws = (float*)d_ws;
    float* xd = ws;              // B*T*256 = 61440 floats
    float* yd = ws + 65536;      // 4096*16 = 65536 floats

    k_downsample<<<BB * TT, 256, 0, stream>>>(x, xd);
    k_ssm_mlp<<<32, 256, 0, stream>>>(xd, w_in, b_in, ln1_g, ln1_b, Lre, Lim,
                                      B_re, B_im, C_re, C_im, Dv, log_step,
                                      ln2_g, ln2_b, w_enc, w_dec, w_out, b_out, yd);
    const int total4 = BB * CH * HH * (WW / 4);
    k_upsample<<<(total4 + 255) / 256, 256, 0, stream>>>(yd, (float*)d_out);
}